// PDEDenseLayer_13056700579985
// MI455X (gfx1250) — compile-verified
//
#include <hip/hip_runtime.h>

// ---------------- problem constants (mirror the reference) ----------------
#define BS      8
#define NID     2
#define NB      (BS * NID)          // 16 independent problems
#define N0      16
#define N1      16
#define G       256                 // N0*N1
#define ORDER   2
#define NORD    5                   // 1 + 2*ORDER
#define NVAR    1280                // G*NORD
#define NIV     16
#define M_REAL  1232                // G + NIV + 480 + 480
#define M_PAD   1248                // multiple of 32 (zero-padded rows)
#define RIDGE   1e-6f
#define BAND    128                 // true half-bandwidth of AtA is 80

// GEMM tiling: each wave computes a 32x32 output region (2x2 WMMA tiles)
#define TILES_1D        (NVAR / 16)                  // 80
#define WTILES_1D       (TILES_1D / 2)               // 40 wave-jobs per dim
#define WAVES_PER_BLK   8                            // 256 threads
#define BLKS_PER_PROB   (WTILES_1D * WTILES_1D / WAVES_PER_BLK)  // 200

typedef __attribute__((ext_vector_type(2)))  float    v2f;
typedef __attribute__((ext_vector_type(8)))  float    v8f;

__device__ __forceinline__ int oidx(int d, int j) {
    return (j == 0) ? 0 : (1 + d * ORDER + (j - 1));
}

// ---------------- kernel 1: assemble sparse A (dense storage) ----------------
__global__ __launch_bounds__(256)
void build_A(const float* __restrict__ coeffs,
             const float* __restrict__ steps0,
             const float* __restrict__ steps1,
             float* __restrict__ A) {
    const int p   = blockIdx.x;
    const int tid = threadIdx.x;
    float* Ab = A + (size_t)p * M_PAD * NVAR;

    // zero the whole padded matrix (vectorized)
    float4* Az = (float4*)Ab;
    const size_t nz = (size_t)M_PAD * NVAR / 4;
    float4 z4 = make_float4(0.f, 0.f, 0.f, 0.f);
    for (size_t i = tid; i < nz; i += 256) Az[i] = z4;
    __syncthreads();

    // eq rows: A[g, g*5+o] = coeffs[p,g,o]
    const float* cb = coeffs + (size_t)p * G * NORD;
    for (int e = tid; e < G * NORD; e += 256) {
        int g = e / NORD;
        Ab[(size_t)g * NVAR + e] = cb[e];
    }
    // iv rows: A[G+i, i*5] = 1
    if (tid < NIV)
        Ab[(size_t)(G + tid) * NVAR + tid * NORD] = 1.0f;

    // derivative rows: 480 rows per coordinate, items indexed by (d, t, k)
    const float fact[3] = {1.f, 1.f, 2.f};
    for (int w = tid; w < 960; w += 256) {
        const int d = w / 480;
        const int q = w % 480;
        const int t = q >> 1;
        const int k = q & 1;
        int gc, gn, row;
        float s;
        if (d == 0) {
            int i0 = t / N1, i1 = t % N1;
            gc = i0 * N1 + i1; gn = gc + N1;
            s  = steps0[p * (N0 - 1) + i0];
            row = G + NIV + t * ORDER + k;
        } else {
            int i0 = t / (N1 - 1), i1 = t % (N1 - 1);
            gc = i0 * N1 + i1; gn = gc + 1;
            s  = steps1[p * (N1 - 1) + i1];
            row = G + NIV + 480 + t * ORDER + k;
        }
        float* Ar = Ab + (size_t)row * NVAR;
        Ar[gn * NORD + oidx(d, k)] = 1.0f;
        float sp = 1.0f;                       // s^(j-k)
        for (int j = k; j <= ORDER; ++j) {
            Ar[gc * NORD + oidx(d, j)] = -sp / fact[j - k];
            sp *= s;
        }
    }
}

// ---------------- kernel 2: AtA = A^T A + eps*I via WMMA (2x2 reg-blocked) ----------------
__global__ __launch_bounds__(256)
void ata_wmma(const float* __restrict__ A, float* __restrict__ C) {
    const int p    = blockIdx.x / BLKS_PER_PROB;
    const int bt   = blockIdx.x % BLKS_PER_PROB;
    const int wave = threadIdx.x >> 5;
    const int lane = threadIdx.x & 31;
    const int t    = bt * WAVES_PER_BLK + wave;          // 0..1599
    const int ti0  = (t / WTILES_1D) * 2;                // 16-row tile index
    const int tj0  = (t % WTILES_1D) * 2;                // 16-col tile index

    const float* Ab = A + (size_t)p * M_PAD * NVAR;
    float*       Cb = C + (size_t)p * NVAR * NVAR;

    const int m  = lane & 15;       // M (A operand) / N (B operand) in tile
    const int h  = lane >> 4;       // lane group selects K pair {0,1} or {2,3}
    const int kh = h * 2;

    // four independent accumulator chains -> hides WMMA latency
    v8f acc00 = {}, acc01 = {}, acc10 = {}, acc11 = {};

    const int ia0 = ti0 * 16 + m;
    const int ia1 = ia0 + 16;
    const int jb0 = tj0 * 16 + m;
    const int jb1 = jb0 + 16;

    for (int kb = 0; kb < M_PAD / 4; ++kb) {
        const float* pa = Ab + (size_t)(kb * 4 + kh) * NVAR;
        if (kb + 8 < M_PAD / 4)
            __builtin_prefetch(pa + (size_t)32 * NVAR + ia0, 0, 1);
        v2f a0, a1, b0, b1;
        a0.x = pa[ia0];        a0.y = pa[NVAR + ia0];
        a1.x = pa[ia1];        a1.y = pa[NVAR + ia1];
        b0.x = pa[jb0];        b0.y = pa[NVAR + jb0];
        b1.x = pa[jb1];        b1.y = pa[NVAR + jb1];
        acc00 = __builtin_amdgcn_wmma_f32_16x16x4_f32(
                    false, a0, false, b0, (short)0, acc00, false, false);
        acc01 = __builtin_amdgcn_wmma_f32_16x16x4_f32(
                    false, a0, false, b1, (short)0, acc01, false, false);
        acc10 = __builtin_amdgcn_wmma_f32_16x16x4_f32(
                    false, a1, false, b0, (short)0, acc10, false, false);
        acc11 = __builtin_amdgcn_wmma_f32_16x16x4_f32(
                    false, a1, false, b1, (short)0, acc11, false, false);
    }

    // C/D layout: VGPR v holds M = v + 8*(lane>>4), N = lane&15
    const int rbase = ti0 * 16 + 8 * h;
    const int cbase = tj0 * 16 + m;
    #pragma unroll
    for (int ii = 0; ii < 2; ++ii) {
        #pragma unroll
        for (int jj = 0; jj < 2; ++jj) {
            const v8f accv = ii == 0 ? (jj == 0 ? acc00 : acc01)
                                     : (jj == 0 ? acc10 : acc11);
            const int col = cbase + jj * 16;
            #pragma unroll
            for (int v = 0; v < 8; ++v) {
                const int row = rbase + ii * 16 + v;
                float cv = accv[v];
                if (row == col) cv += RIDGE;
                Cb[(size_t)row * NVAR + col] = cv;
            }
        }
    }
}

// ---------------- kernel 3: Atb (b nonzero only in first 272 rows) ----------------
__global__ __launch_bounds__(256)
void atb_kernel(const float* __restrict__ A,
                const float* __restrict__ rhs,
                const float* __restrict__ iv,
                float* __restrict__ Atb) {
    const int p = blockIdx.x;
    const float* Ab = A   + (size_t)p * M_PAD * NVAR;
    const float* rb = rhs + (size_t)p * G;
    const float* ib = iv  + (size_t)p * NIV;
    for (int col = threadIdx.x; col < NVAR; col += 256) {
        float acc = 0.f;
        for (int k = 0; k < G; ++k)
            acc += Ab[(size_t)k * NVAR + col] * rb[k];
        for (int k = 0; k < NIV; ++k)
            acc += Ab[(size_t)(G + k) * NVAR + col] * ib[k];
        Atb[(size_t)p * NVAR + col] = acc;
    }
}

// ---------------- kernel 4: band-aware elimination + wave32 back-sub ----------------
__global__ __launch_bounds__(1024)
void solve_kernel(float* __restrict__ C,
                  const float* __restrict__ Atb,
                  float* __restrict__ X) {
    const int p    = blockIdx.x;
    const int tid  = threadIdx.x;
    const int lane = tid & 31;
    float* Cb = C + (size_t)p * NVAR * NVAR;

    __shared__ float ys[NVAR];
    __shared__ float xs[NVAR];

    for (int i = tid; i < NVAR; i += 1024) ys[i] = Atb[(size_t)p * NVAR + i];
    __syncthreads();

    // forward elimination (no pivoting: SPD + ridge). Band structure means
    // multipliers outside |r-c|<=84 are exactly zero -> truncate at BAND.
    for (int c = 0; c < NVAR; ++c) {
        const float pinv = 1.0f / Cb[(size_t)c * NVAR + c];
        const float yc   = ys[c];
        const int   rend = (c + BAND < NVAR) ? (c + BAND) : NVAR;
        const int   jend = rend;
        const float* src = Cb + (size_t)c * NVAR;
        for (int r = c + 1 + tid; r < rend; r += 1024) {
            const float f = Cb[(size_t)r * NVAR + c] * pinv;
            if (f != 0.0f) {
                float* dst = Cb + (size_t)r * NVAR;
                for (int j = c + 1; j < jend; ++j) dst[j] -= f * src[j];
                ys[r] -= f * yc;
            }
        }
        __syncthreads();
    }

    // back substitution by wave 0 (banded: j-c <= 84 < BAND)
    if (tid < 32) {
        for (int c = NVAR - 1; c >= 0; --c) {
            const int jend = (c + BAND < NVAR) ? (c + BAND) : NVAR;
            const float* src = Cb + (size_t)c * NVAR;
            float part = 0.f;
            for (int j = c + 1 + lane; j < jend; j += 32) part += src[j] * xs[j];
            for (int off = 16; off > 0; off >>= 1)
                part += __shfl_down(part, (unsigned)off, 32);
            if (lane == 0) xs[c] = (ys[c] - part) / src[c];
        }
    }
    __syncthreads();
    for (int i = tid; i < NVAR; i += 1024) X[(size_t)p * NVAR + i] = xs[i];
}

// ---------------- kernel 5: scatter (u0, u) into d_out ----------------
__global__ __launch_bounds__(256)
void scatter_kernel(const float* __restrict__ X, float* __restrict__ out) {
    const int idx = blockIdx.x * 256 + threadIdx.x;
    if (idx >= NB * NVAR) return;
    const int p = idx / NVAR;
    const int i = idx % NVAR;
    const float v = X[idx];
    out[NB * G + idx] = v;                          // u  (after u0 block)
    if (i % NORD == 0) out[p * G + i / NORD] = v;   // u0
}

// ---------------- host launch ----------------
extern "C" void kernel_launch(void* const* d_in, const int* in_sizes, int n_in,
                              void* d_out, int out_size, void* d_ws, size_t ws_size,
                              hipStream_t stream) {
    const float* coeffs = (const float*)d_in[0];  // (8,2,256,5)
    const float* rhs    = (const float*)d_in[1];  // (8,2,256)
    const float* iv     = (const float*)d_in[2];  // (8,2,16)
    const float* s0     = (const float*)d_in[3];  // (8,2,15)
    const float* s1     = (const float*)d_in[4];  // (8,2,15)
    float* out = (float*)d_out;                   // 4096 (u0) + 20480 (u)

    float* ws  = (float*)d_ws;
    float* A   = ws;                                        // NB*M_PAD*NVAR
    float* AtA = A   + (size_t)NB * M_PAD * NVAR;           // NB*NVAR*NVAR
    float* Atb = AtA + (size_t)NB * NVAR * NVAR;            // NB*NVAR
    float* X   = Atb + (size_t)NB * NVAR;                   // NB*NVAR

    build_A<<<NB, 256, 0, stream>>>(coeffs, s0, s1, A);
    ata_wmma<<<NB * BLKS_PER_PROB, 256, 0, stream>>>(A, AtA);
    atb_kernel<<<NB, 256, 0, stream>>>(A, rhs, iv, Atb);
    solve_kernel<<<NB, 1024, 0, stream>>>(AtA, Atb, X);
    scatter_kernel<<<(NB * NVAR + 255) / 256, 256, 0, stream>>>(X, out);
}